// SmoothTransitionLoss_70205535421041
// MI455X (gfx1250) — compile-verified
//
#include <hip/hip_runtime.h>
#include <hip/hip_bf16.h>

typedef __attribute__((ext_vector_type(2))) float v2f;
typedef __attribute__((ext_vector_type(8))) float v8f;

#define KSZ    11
#define IMG_H  1024
#define IMG_W  1024
#define BATCH  8
#define CHAN   3
#define TILE   16
#define WAVES  4

// Per-wave LDS slice layout (float indices; all bases/strides even -> 8B aligned)
#define IMG_ST   40
#define IMG_ROWS 38
#define IMG_OFF  0
#define IMG_SZ   (IMG_ROWS*IMG_ST)   // 1520
#define CA_ST    40
#define CA_ROWS  28
#define CA_OFF   (IMG_OFF+IMG_SZ)    // 1520
#define CA_SZ    (CA_ROWS*CA_ST)     // 1120
#define SM_ST    28
#define SM_OFF   (CA_OFF+CA_SZ)      // 2640
#define SM_SZ    (26*SM_ST)          // 728
#define CV_ST    28
#define CV_OFF   (SM_OFF+SM_SZ)      // 3368
#define CV_SZ    (16*CV_ST)          // 448
#define SLICE    (CV_OFF+CV_SZ)      // 3816 floats = 15264 B

__device__ __forceinline__ void cfence() { asm volatile("" ::: "memory"); }

// Column blur: O[m][n] = sum_v g[v-m] * S[v0+v][x0+n]; A = banded const, B = data.
__device__ __forceinline__ v8f wmma_colpass(const float* S, int stride, int v0, int x0,
                                            const v2f* gb, int lane) {
  const int n  = lane & 15;
  const int j0 = (lane < 16) ? 0 : 2;
  v8f acc = {};
#pragma unroll
  for (int c = 0; c < 7; ++c) {
    const float* p = S + (v0 + 4 * c + j0) * stride + x0 + n;
    v2f b; b.x = p[0]; b.y = p[stride];
    acc = __builtin_amdgcn_wmma_f32_16x16x4_f32(false, gb[c], false, b,
                                                (short)0, acc, false, false);
  }
  return acc;
}

// Row blur: O[m][n] = sum_u S[y0+m][u0+u] * g[u-n]; A = data, B = banded const.
__device__ __forceinline__ v8f wmma_rowpass(const float* S, int stride, int y0, int u0,
                                            const v2f* gb, int lane) {
  const int m  = lane & 15;
  const int k0 = (lane < 16) ? 0 : 2;
  v8f acc = {};
#pragma unroll
  for (int c = 0; c < 7; ++c) {
    v2f a = *(const v2f*)(S + (y0 + m) * stride + u0 + 4 * c + k0);
    acc = __builtin_amdgcn_wmma_f32_16x16x4_f32(false, a, false, gb[c],
                                                (short)0, acc, false, false);
  }
  return acc;
}

__device__ __forceinline__ void store_tile(float* D, int stride, int y0, int x0,
                                           v8f t, int lane) {
  const int n  = lane & 15;
  const int hi = (lane < 16) ? 0 : 8;
#pragma unroll
  for (int r = 0; r < 8; ++r) D[(y0 + hi + r) * stride + x0 + n] = t[r];
}

// Per 16x16 tile: smooth (mean) and local variance via WMMA separable blurs.
__device__ __forceinline__ void compute_stats(const float* gsrc, int gy0, int gx0,
                                              float* img, float* ca, float* sm, float* cv,
                                              const v2f* gb, int lane,
                                              v8f* outMean, v8f* outVar) {
  // 36x36 halo load (zero pad = reference conv padding)
  for (int i = lane; i < 36 * 36; i += 32) {
    int p = i / 36, q = i - p * 36;
    int gy = gy0 - 10 + p, gx = gx0 - 10 + q;
    float v = 0.f;
    if ((unsigned)gy < IMG_H && (unsigned)gx < IMG_W) v = gsrc[(size_t)gy * IMG_W + gx];
    img[p * IMG_ST + q] = v;
  }
  cfence();
  // Pass 1: column blur img -> ca  (26 rows x 36 cols of valid data)
#pragma unroll
  for (int p0 = 0; p0 <= 10; p0 += 10) {
    store_tile(ca, CA_ST, p0, 0,  wmma_colpass(img, IMG_ST, p0, 0,  gb, lane), lane);
    store_tile(ca, CA_ST, p0, 16, wmma_colpass(img, IMG_ST, p0, 16, gb, lane), lane);
    store_tile(ca, CA_ST, p0, 20, wmma_colpass(img, IMG_ST, p0, 20, gb, lane), lane);
  }
  cfence();
  // Pass 2: row blur ca -> sm (26x26 smooth)
#pragma unroll
  for (int y0 = 0; y0 <= 10; y0 += 10)
#pragma unroll
    for (int x0 = 0; x0 <= 10; x0 += 10)
      store_tile(sm, SM_ST, y0, x0, wmma_rowpass(ca, CA_ST, y0, x0, gb, lane), lane);
  cfence();
  // r2 = (img - smooth)^2 on 26x26 (reuse ca; rows 26,27 stay zero)
  for (int i = lane; i < 26 * 26; i += 32) {
    int p = i / 26, q = i - p * 26;
    float d = img[(p + 5) * IMG_ST + (q + 5)] - sm[p * SM_ST + q];
    ca[p * CA_ST + q] = d * d;
  }
  cfence();
  // Pass 3: column blur r2 -> cv (16x26)
#pragma unroll
  for (int x0 = 0; x0 <= 10; x0 += 10)
    store_tile(cv, CV_ST, 0, x0, wmma_colpass(ca, CA_ST, 0, x0, gb, lane), lane);
  cfence();
  // Pass 4: row blur cv -> var (register tile)
  v8f var = wmma_rowpass(cv, CV_ST, 0, 0, gb, lane);
  const int n = lane & 15, hi = (lane < 16) ? 0 : 8;
  v8f mean;
#pragma unroll
  for (int r = 0; r < 8; ++r) mean[r] = sm[(hi + r + 5) * SM_ST + (n + 5)];
  *outMean = mean;
  *outVar  = var;
}

__global__ void stl_init(const float* __restrict__ k2, float* ws) {
  int t = threadIdx.x;
  if (t < KSZ) {                      // separable 1-D kernel: row sums of normalized 2-D
    float s = 0.f;
    for (int j = 0; j < KSZ; ++j) s += k2[t * KSZ + j];
    ws[2 + t] = s;
  }
  if (t == 11) ws[0] = 0.f;
  if (t == 12) ws[1] = 0.f;
}

__global__ __launch_bounds__(WAVES * 32, 1) void stl_main(
    const float* __restrict__ pred, const float* __restrict__ target,
    const float* __restrict__ mask, float* ws) {
  __shared__ __align__(16) float lds[WAVES * SLICE];
  const int lane = threadIdx.x & 31;
  const int wave = threadIdx.x >> 5;
  float* S   = lds + wave * SLICE;
  float* img = S + IMG_OFF;
  float* ca  = S + CA_OFF;
  float* sm  = S + SM_OFF;
  float* cv  = S + CV_OFF;

  const int b   = blockIdx.z;
  const int gy0 = blockIdx.y * TILE;
  const int gx0 = (blockIdx.x * WAVES + wave) * TILE;

  for (int i = lane; i < SLICE; i += 32) S[i] = 0.f;   // zero padding regions once
  cfence();

  float g[KSZ];
#pragma unroll
  for (int i = 0; i < KSZ; ++i) g[i] = ws[2 + i];
  const int idx = lane & 15;
  const int j0  = (lane < 16) ? 0 : 2;
  v2f gb[7];                                           // banded conv matrix, per lane
#pragma unroll
  for (int c = 0; c < 7; ++c) {
    int i0 = 4 * c + j0 - idx;
    gb[c].x = (i0 >= 0 && i0 < KSZ) ? g[i0] : 0.f;
    gb[c].y = (i0 + 1 >= 0 && i0 + 1 < KSZ) ? g[i0 + 1] : 0.f;
  }

  // ---- boundary mask: separable 11x11 max-pool minus center ----
  const float* mplane = mask + (size_t)b * IMG_H * IMG_W;
  for (int i = lane; i < 26 * 26; i += 32) {
    int p = i / 26, q = i - p * 26;
    int gy = gy0 - 5 + p, gx = gx0 - 5 + q;
    float v = 0.f;                                     // 0-pad == -inf pad for {0,1} mask
    if ((unsigned)gy < IMG_H && (unsigned)gx < IMG_W) v = mplane[(size_t)gy * IMG_W + gx];
    ca[p * CA_ST + q] = v;
  }
  cfence();
  for (int i = lane; i < 16 * 26; i += 32) {           // column max
    int y = i / 26, q = i - y * 26;
    float mx = ca[y * CA_ST + q];
#pragma unroll
    for (int t = 1; t < KSZ; ++t) mx = fmaxf(mx, ca[(y + t) * CA_ST + q]);
    sm[y * SM_ST + q] = mx;
  }
  cfence();
  const int n = lane & 15, hi = (lane < 16) ? 0 : 8;
  v8f bm;
#pragma unroll
  for (int r = 0; r < 8; ++r) {                        // row max - center (C/D layout)
    int y = hi + r;
    float mx = sm[y * SM_ST + n];
#pragma unroll
    for (int s = 1; s < KSZ; ++s) mx = fmaxf(mx, sm[y * SM_ST + n + s]);
    bm[r] = mx - ca[(y + 5) * CA_ST + (n + 5)];
  }
  cfence();

  // ---- channels ----
  float accM = 0.f, accV = 0.f;
  for (int c = 0; c < CHAN; ++c) {
    const size_t plane = ((size_t)b * CHAN + c) * (size_t)IMG_H * IMG_W;
    v8f pm, pv, tm, tv;
    compute_stats(pred + plane,   gy0, gx0, img, ca, sm, cv, gb, lane, &pm, &pv);
    cfence();
    compute_stats(target + plane, gy0, gx0, img, ca, sm, cv, gb, lane, &tm, &tv);
#pragma unroll
    for (int r = 0; r < 8; ++r) {
      accM += bm[r] * fabsf(pm[r] - tm[r]);            // bm in {0,1}: == |a*bm - b*bm|
      accV += bm[r] * fabsf(pv[r] - tv[r]);
    }
    cfence();
  }

#pragma unroll
  for (int off = 16; off > 0; off >>= 1) {
    accM += __shfl_xor(accM, off, 32);
    accV += __shfl_xor(accV, off, 32);
  }
  if (lane == 0) {
    unsafeAtomicAdd(&ws[0], accM);
    unsafeAtomicAdd(&ws[1], accV);
  }
}

__global__ void stl_finalize(const float* __restrict__ ws, float* out) {
  const float invN = 1.0f / (float)((size_t)BATCH * CHAN * IMG_H * IMG_W);
  out[0] = fminf(ws[0] * invN + 0.5f * ws[1] * invN, 3.0f);
}

extern "C" void kernel_launch(void* const* d_in, const int* in_sizes, int n_in,
                              void* d_out, int out_size, void* d_ws, size_t ws_size,
                              hipStream_t stream) {
  const float* pred   = (const float*)d_in[0];
  const float* target = (const float*)d_in[1];
  const float* mask   = (const float*)d_in[2];
  const float* k2     = (const float*)d_in[3];
  float* ws  = (float*)d_ws;   // [0]=sumMean [1]=sumVar [2..12]=g
  float* out = (float*)d_out;

  stl_init<<<1, 32, 0, stream>>>(k2, ws);
  dim3 grid(64 / WAVES, 64, BATCH);
  stl_main<<<grid, WAVES * 32, 0, stream>>>(pred, target, mask, ws);
  stl_finalize<<<1, 1, 0, stream>>>(ws, out);
}